// LinearEncoder_57071525429452
// MI455X (gfx1250) — compile-verified
//
#include <hip/hip_runtime.h>
#include <hip/hip_bf16.h>

typedef __attribute__((ext_vector_type(2))) float v2f;
typedef __attribute__((ext_vector_type(8))) float v8f;

#define C_DIM 32

// ---------------------------------------------------------------------------
// Kernel 1: init degree to 1.0 (self loop)
// ---------------------------------------------------------------------------
__global__ void gcn_init_deg(float* __restrict__ deg, int n) {
    int i = blockIdx.x * blockDim.x + threadIdx.x;
    if (i < n) deg[i] = 1.0f;
}

// ---------------------------------------------------------------------------
// Kernel 2: degree count over targets
// ---------------------------------------------------------------------------
__global__ void gcn_degree(const int* __restrict__ col, float* __restrict__ deg, int E) {
    int e = blockIdx.x * blockDim.x + threadIdx.x;
    if (e < E) atomicAdd(&deg[col[e]], 1.0f);
}

// ---------------------------------------------------------------------------
// Kernel 3: deg -> deg^{-1/2} in place (deg >= 1 always, self-loops included)
// ---------------------------------------------------------------------------
__global__ void gcn_rsqrt(float* __restrict__ deg, int n) {
    int i = blockIdx.x * blockDim.x + threadIdx.x;
    if (i < n) deg[i] = rsqrtf(deg[i]);
}

// ---------------------------------------------------------------------------
// Kernel 4: h = x @ W^T via V_WMMA_F32_16X16X4_F32 (exact f32 math).
// One wave computes a 16-row tile of h (16 x 32), two 16x16 N-tiles,
// chaining 8 K=4 WMMAs per accumulator.
//
// Layouts (ISA 7.12.2, 32-bit):
//   A 16x4 : lane l holds A[l&15][2*(l>>4)+{0,1}] in 2 VGPRs
//   B 4x16 : lane l holds B[2*(l>>4)+{0,1}][l&15];  B[k][n] = W[n][k]
//   C 16x16: VGPR r holds D[r + 8*(l>>4)][l&15]
// ---------------------------------------------------------------------------
__global__ void gcn_linear_wmma(const float* __restrict__ x,
                                const float* __restrict__ W,
                                float* __restrict__ h, int n_tiles) {
    const int lane = threadIdx.x & 31;
    const int wave = threadIdx.x >> 5;
    const int tile = blockIdx.x * (blockDim.x >> 5) + wave;
    if (tile >= n_tiles) return;             // wave-uniform guard: EXEC all-1s inside

    const int half = lane >> 4;              // 0 or 1
    const int lid  = lane & 15;
    const int arow = tile * 16 + lid;        // A-row owned by this lane
    const int n0   = lid;                    // B cols 0..15
    const int n1   = 16 + lid;               // B cols 16..31

    v8f acc0 = {};
    v8f acc1 = {};

#pragma unroll
    for (int kk = 0; kk < C_DIM; kk += 4) {
        const int k0 = kk + 2 * half;        // even -> 8B aligned float2 loads
        const v2f a  = *(const v2f*)(x + arow * C_DIM + k0);
        const v2f b0 = *(const v2f*)(W + n0 * C_DIM + k0);   // W[n][k0], W[n][k0+1]
        const v2f b1 = *(const v2f*)(W + n1 * C_DIM + k0);
        acc0 = __builtin_amdgcn_wmma_f32_16x16x4_f32(
                   /*neg_a=*/false, a, /*neg_b=*/false, b0,
                   /*c_mod=*/(short)0, acc0, /*reuse_a=*/false, /*reuse_b=*/false);
        acc1 = __builtin_amdgcn_wmma_f32_16x16x4_f32(
                   false, a, false, b1, (short)0, acc1, false, false);
    }

#pragma unroll
    for (int r = 0; r < 8; ++r) {
        const int m = tile * 16 + r + 8 * half;
        h[m * C_DIM + lid]      = acc0[r];
        h[m * C_DIM + 16 + lid] = acc1[r];
    }
}

// Scalar remainder path (only launched if N % 16 != 0; N=100000 -> never).
__global__ void gcn_linear_rem(const float* __restrict__ x,
                               const float* __restrict__ W,
                               float* __restrict__ h, int start, int n) {
    int t = blockIdx.x * blockDim.x + threadIdx.x;
    int i = start + (t >> 5);
    int c = t & 31;
    if (i >= n) return;
    float acc = 0.f;
#pragma unroll
    for (int k = 0; k < C_DIM; ++k) acc += x[i * C_DIM + k] * W[c * C_DIM + k];
    h[i * C_DIM + c] = acc;
}

// ---------------------------------------------------------------------------
// Kernel 5: out[i][c] = b[c] + h[i][c] * dinv[i]^2   (bias + self-loop term)
// ---------------------------------------------------------------------------
__global__ void gcn_init_out(const float* __restrict__ h,
                             const float* __restrict__ dinv,
                             const float* __restrict__ bias,
                             float* __restrict__ out, int n) {
    int t = blockIdx.x * blockDim.x + threadIdx.x;
    if (t >= n * C_DIM) return;
    int i = t >> 5;
    int c = t & 31;
    float d = dinv[i];
    out[t] = bias[c] + h[t] * d * d;
}

// ---------------------------------------------------------------------------
// Kernel 6: edge scatter. One wave (32 lanes) per edge: lane = channel.
// Index / dinv loads are wave-uniform; h/out accesses fully coalesced.
// out lives in L2 (12.8 MB << 192 MB), so atomics are L2-resident.
// ---------------------------------------------------------------------------
__global__ void gcn_scatter(const float* __restrict__ h,
                            const int* __restrict__ row,
                            const int* __restrict__ col,
                            const float* __restrict__ dinv,
                            float* __restrict__ out, int E) {
    int t = blockIdx.x * blockDim.x + threadIdx.x;
    int e = t >> 5;
    int c = t & 31;
    if (e >= E) return;
    int r  = row[e];
    int cl = col[e];
    float nrm = dinv[r] * dinv[cl];
    atomicAdd(&out[cl * C_DIM + c], h[r * C_DIM + c] * nrm);
}

// ---------------------------------------------------------------------------
extern "C" void kernel_launch(void* const* d_in, const int* in_sizes, int n_in,
                              void* d_out, int out_size, void* d_ws, size_t ws_size,
                              hipStream_t stream) {
    const float* x  = (const float*)d_in[0];
    const int*   ei = (const int*)d_in[1];
    const float* W  = (const float*)d_in[2];
    const float* b  = (const float*)d_in[3];
    float*       out = (float*)d_out;

    const int N = in_sizes[0] / C_DIM;   // 100000
    const int E = in_sizes[1] / 2;       // 1600000
    const int* row = ei;                 // sources
    const int* col = ei + E;             // targets

    // Workspace layout: dinv [N floats] | h [N*32 floats]
    float* dinv = (float*)d_ws;
    float* h    = dinv + ((N + 63) & ~63);

    const int B = 256;

    // 1) deg = 1 (self loop)
    gcn_init_deg<<<(N + B - 1) / B, B, 0, stream>>>(dinv, N);
    // 2) deg += 1 per incoming edge
    gcn_degree<<<(E + B - 1) / B, B, 0, stream>>>(col, dinv, E);
    // 3) dinv = deg^{-1/2}
    gcn_rsqrt<<<(N + B - 1) / B, B, 0, stream>>>(dinv, N);

    // 4) h = x @ W^T  (WMMA f32 16x16x4; 8 waves/block -> 128 rows/block)
    const int n_tiles = N / 16;
    const int waves_per_block = B / 32;
    if (n_tiles > 0)
        gcn_linear_wmma<<<(n_tiles + waves_per_block - 1) / waves_per_block, B, 0, stream>>>(
            x, W, h, n_tiles);
    const int rem_start = n_tiles * 16;
    if (rem_start < N) {
        const int rem_threads = (N - rem_start) * C_DIM;
        gcn_linear_rem<<<(rem_threads + B - 1) / B, B, 0, stream>>>(x, W, h, rem_start, N);
    }

    // 5) out = b + h * dinv^2  (self-loop message)
    {
        long long tot = (long long)N * C_DIM;
        gcn_init_out<<<(int)((tot + B - 1) / B), B, 0, stream>>>(h, dinv, b, out, N);
    }

    // 6) scatter-add edge messages
    {
        long long tot = (long long)E * C_DIM;
        gcn_scatter<<<(int)((tot + B - 1) / B), B, 0, stream>>>(h, row, col, dinv, out, E);
    }
}